// RoiAlign_257698038263
// MI455X (gfx1250) — compile-verified
//
#include <hip/hip_runtime.h>
#include <cstdint>

// ---------------------------------------------------------------------------
// ROI Align (FPN, 7x7, C=256) for gfx1250 using the Tensor Data Mover.
// Memory-bound op: ~0.5-1 GB of traffic -> ~40-50us at 23.3 TB/s. No matmul
// structure exists (per-sample corners differ), so the CDNA5 path used here is
// TDM async tile loads to LDS (TENSORcnt double buffering), not WMMA.
// ---------------------------------------------------------------------------

#define POOL   7
#define NSAMP  49          // 7*7
#define CCH    256         // channels
#define TILE_F 1024        // 2 rows * 2 pixels * 256 ch floats per sample tile
#define NWAVE  8

typedef unsigned int u32x4 __attribute__((ext_vector_type(4)));
typedef int          i32x8 __attribute__((ext_vector_type(8)));
typedef int          i32x4 __attribute__((ext_vector_type(4)));

#ifndef __has_builtin
#define __has_builtin(x) 0
#endif

#if defined(__HIP_DEVICE_COMPILE__) && \
    __has_builtin(__builtin_amdgcn_tensor_load_to_lds) && \
    __has_builtin(__builtin_amdgcn_s_wait_tensorcnt)
#define USE_TDM 1
#else
#define USE_TDM 0
#endif

// Sample-point geometry (matches reference: unclamped floor; weights from it).
__device__ __forceinline__ void sample_geom(float by1, float bx1, float by2, float bx2,
                                            int H, int s,
                                            int& y0, int& x0, float& wy, float& wx) {
    const int py = s / POOL;
    const int px = s - py * POOL;
    const float fy = (float)py / 6.0f;
    const float fx = (float)px / 6.0f;
    const float ys = (by1 + fy * (by2 - by1)) * (float)(H - 1);
    const float xs = (bx1 + fx * (bx2 - bx1)) * (float)(H - 1);
    const float y0f = floorf(ys);
    const float x0f = floorf(xs);
    y0 = (int)y0f;
    x0 = (int)x0f;
    wy = ys - y0f;
    wx = xs - x0f;
}

__device__ __forceinline__ float4 blend4(float4 tl, float4 tr, float4 bl, float4 br,
                                         float wx, float wy) {
    float4 o;
    float t, b;
    t = tl.x + (tr.x - tl.x) * wx; b = bl.x + (br.x - bl.x) * wx; o.x = t + (b - t) * wy;
    t = tl.y + (tr.y - tl.y) * wx; b = bl.y + (br.y - bl.y) * wx; o.y = t + (b - t) * wy;
    t = tl.z + (tr.z - tl.z) * wx; b = bl.z + (br.z - bl.z) * wx; o.z = t + (b - t) * wy;
    t = tl.w + (tr.w - tl.w) * wx; b = bl.w + (br.w - bl.w) * wx; o.w = t + (b - t) * wy;
    return o;
}

#if USE_TDM
// Issue a TDM 2D tile load: 2 rows x 512 elements (2 pixels x 256 ch, f32),
// row stride = W*C elements. tensor_dim0/1 are set to the remaining extent from
// the tile start so OOB pixels/rows zero-fill (those always have weight 0).
__device__ __forceinline__ void tdm_issue(const float* tile_ptr, unsigned lds_off,
                                          int rem_x_elems, int rem_rows,
                                          int row_stride_elems) {
    const unsigned long long ga = (unsigned long long)(uintptr_t)tile_ptr;
    u32x4 g0;
    g0.s0 = 1u;                                       // count = 1 (valid D#)
    g0.s1 = lds_off;                                  // LDS byte address
    g0.s2 = (unsigned)ga;                             // global_addr[31:0]
    g0.s3 = (unsigned)((ga >> 32) & 0x01FFFFFFu)      // global_addr[56:32]
          | (2u << 30);                               // type = 2 ("image")
    const unsigned td0 = (unsigned)rem_x_elems;       // tensor_dim0 (elements)
    const unsigned td1 = (unsigned)rem_rows;          // tensor_dim1 (rows)
    const unsigned st0 = (unsigned)row_stride_elems;  // tensor_dim0_stride
    i32x8 g1;
    g1.s0 = (int)(2u << 16);                          // data_size = 2 -> 4B; no mask/pad
    g1.s1 = (int)((td0 & 0xFFFFu) << 16);             // tensor_dim0[15:0] @ [63:48]
    g1.s2 = (int)((td0 >> 16) | ((td1 & 0xFFFFu) << 16)); // dim0 hi | dim1 lo
    g1.s3 = (int)((td1 >> 16) | (512u << 16));        // dim1 hi | tile_dim0 = 512
    g1.s4 = 2;                                        // tile_dim1 = 2, tile_dim2 = 0
    g1.s5 = (int)st0;                                 // tensor_dim0_stride[31:0]
    g1.s6 = 0;                                        // stride hi | dim1_stride lo (unused 2D)
    g1.s7 = 0;
    const i32x4 gz4 = {0, 0, 0, 0};                   // groups 2/3 unused (2D tensor)
    const i32x8 gz8 = {0, 0, 0, 0, 0, 0, 0, 0};       // extra operand (clang-23 6-arg form)
    __builtin_amdgcn_tensor_load_to_lds(g0, g1, gz4, gz4, gz8, 0);
}
#endif

__global__ __launch_bounds__(256, 1)
void roi_align_tdm_kernel(const float* __restrict__ boxes,
                          const float* __restrict__ f2, const float* __restrict__ f3,
                          const float* __restrict__ f4, const float* __restrict__ f5,
                          float* __restrict__ out, int N /* rois per batch */) {
    __shared__ float lds[NWAVE][2][TILE_F];   // 64 KB: per-wave double buffer

    const int roi  = blockIdx.x;
    const int b    = roi / N;
    const int tid  = threadIdx.x;
    const int wid  = tid >> 5;                // wave id (wave32)
    const int lane = tid & 31;

    // Box (uniform across block -> scalar loads).
    const float by1 = boxes[roi * 4 + 0];
    const float bx1 = boxes[roi * 4 + 1];
    const float by2 = boxes[roi * 4 + 2];
    const float bx2 = boxes[roi * 4 + 3];

    // Level selection (reference: clip(round(log2(sqrt(max(h*w,1e-12))*1024/224))+4, 2, 5)).
    const float h = by2 - by1, w = bx2 - bx1;
    const float scale = sqrtf(fmaxf(h * w, 1e-12f)) * (float)(1024.0 / 224.0);
    const float lf = rintf(log2f(scale)) + 4.0f;
    const int lvl = (int)fminf(fmaxf(lf, 2.0f), 5.0f);

    const float* feat; int H;
    if (lvl <= 2)      { feat = f2; H = 256; }
    else if (lvl == 3) { feat = f3; H = 128; }
    else if (lvl == 4) { feat = f4; H = 64;  }
    else               { feat = f5; H = 32;  }

    const float* fb    = feat + (size_t)b * (size_t)H * (size_t)H * CCH;
    float*       obase = out  + (size_t)roi * NSAMP * CCH;
    const int    c0    = lane * 8;            // 8 channels per lane (2 x float4)

#if USE_TDM
    const unsigned lbuf0 = (unsigned)(uintptr_t)&lds[wid][0][0];
    const unsigned lbuf1 = (unsigned)(uintptr_t)&lds[wid][1][0];
    const int n = (NSAMP - wid + NWAVE - 1) / NWAVE;   // samples owned by this wave

    // Prologue: issue sample 0 into buffer 0.
    {
        int y0, x0; float wy, wx;
        sample_geom(by1, bx1, by2, bx2, H, wid, y0, x0, wy, wx);
        tdm_issue(fb + ((size_t)y0 * H + x0) * CCH, lbuf0,
                  (H - x0) * CCH, H - y0, H * CCH);
    }

    for (int i = 0; i < n; ++i) {
        const int s = wid + i * NWAVE;
        int y0, x0; float wy, wx;
        sample_geom(by1, bx1, by2, bx2, H, s, y0, x0, wy, wx);

        if (i + 1 < n) {
            // Prefetch next sample's tile into the other buffer.
            const int s2 = s + NWAVE;
            int yy0, xx0; float wwy, wwx;
            sample_geom(by1, bx1, by2, bx2, H, s2, yy0, xx0, wwy, wwx);
            tdm_issue(fb + ((size_t)yy0 * H + xx0) * CCH,
                      (i & 1) ? lbuf0 : lbuf1,
                      (H - xx0) * CCH, H - yy0, H * CCH);
            __builtin_amdgcn_s_wait_tensorcnt((short)1);   // oldest (current) tile done
        } else {
            __builtin_amdgcn_s_wait_tensorcnt((short)0);
        }

        const float* buf = (i & 1) ? &lds[wid][1][0] : &lds[wid][0][0];
        // Tile layout in LDS: [tl(256) | tr(256) | bl(256) | br(256)] floats.
        float* op = obase + (size_t)s * CCH;
        #pragma unroll
        for (int k = 0; k < 2; ++k) {
            const int cc = c0 + k * 4;
            const float4 tl = *(const float4*)(buf +   0 + cc);
            const float4 tr = *(const float4*)(buf + 256 + cc);
            const float4 bl = *(const float4*)(buf + 512 + cc);
            const float4 br = *(const float4*)(buf + 768 + cc);
            *(float4*)(op + cc) = blend4(tl, tr, bl, br, wx, wy);
        }
    }
#else
    // Fallback (no TDM builtin): direct clamped global gathers, 8 ch per lane.
    (void)lds;
    for (int s = wid; s < NSAMP; s += NWAVE) {
        int y0, x0; float wy, wx;
        sample_geom(by1, bx1, by2, bx2, H, s, y0, x0, wy, wx);
        const int y0c = min(max(y0, 0), H - 1);
        const int x0c = min(max(x0, 0), H - 1);
        const int y1c = min(y0c + 1, H - 1);
        const int x1c = min(x0c + 1, H - 1);
        const float* p00 = fb + ((size_t)y0c * H + x0c) * CCH;
        const float* p01 = fb + ((size_t)y0c * H + x1c) * CCH;
        const float* p10 = fb + ((size_t)y1c * H + x0c) * CCH;
        const float* p11 = fb + ((size_t)y1c * H + x1c) * CCH;
        float* op = obase + (size_t)s * CCH;
        #pragma unroll
        for (int k = 0; k < 2; ++k) {
            const int cc = c0 + k * 4;
            const float4 tl = *(const float4*)(p00 + cc);
            const float4 tr = *(const float4*)(p01 + cc);
            const float4 bl = *(const float4*)(p10 + cc);
            const float4 br = *(const float4*)(p11 + cc);
            *(float4*)(op + cc) = blend4(tl, tr, bl, br, wx, wy);
        }
    }
#endif
}

extern "C" void kernel_launch(void* const* d_in, const int* in_sizes, int n_in,
                              void* d_out, int out_size, void* d_ws, size_t ws_size,
                              hipStream_t stream) {
    (void)n_in; (void)out_size; (void)d_ws; (void)ws_size;
    const float* boxes = (const float*)d_in[0];
    const float* f2    = (const float*)d_in[1];
    const float* f3    = (const float*)d_in[2];
    const float* f4    = (const float*)d_in[3];
    const float* f5    = (const float*)d_in[4];
    float*       out   = (float*)d_out;

    const int nroi = in_sizes[0] / 4;                 // B*N
    const int B    = in_sizes[1] / (256 * 256 * 256); // from feat2 [B,256,256,256]
    const int N    = nroi / (B > 0 ? B : 1);

    hipLaunchKernelGGL(roi_align_tdm_kernel, dim3(nroi), dim3(256), 0, stream,
                       boxes, f2, f3, f4, f5, out, N);
}